// MultiBoxLoss_48842368090689
// MI455X (gfx1250) — compile-verified
//
#include <hip/hip_runtime.h>
#include <math.h>

#define B_ 64
#define P_ 8732
#define M_ 16
#define C_ 21
#define BG_ 20
#define NTHREADS 256
#define NWAVES (NTHREADS / 32)

#ifndef __has_builtin
#define __has_builtin(x) 0
#endif

#if __has_builtin(__builtin_amdgcn_global_load_async_to_lds_b128)
#define HAVE_ASYNC_LDS 1
#else
#define HAVE_ASYNC_LDS 0
#endif

typedef __attribute__((ext_vector_type(2))) float v2f;
typedef __attribute__((ext_vector_type(8))) float v8f;
typedef int v4i_ld __attribute__((vector_size(16)));  // matches builtin param type

// ---------------------------------------------------------------------------
// Wave(32)-wide sum using V_WMMA_F32_16X16X4_F32 with A = ones:
// D[i][j] = sum_k B[k][j] + C[i][j].  Row-sum of any row of D equals the sum
// of ALL 64 B slots (bijective lane layout), so feeding one float per lane in
// B[0] (B[1]=0) and summing D row entries across 16 lanes yields the wave sum.
// Call only from uniform control flow (EXEC must be all ones).
// ---------------------------------------------------------------------------
__device__ __forceinline__ float wave_sum32(float v) {
#if __has_builtin(__builtin_amdgcn_wmma_f32_16x16x4_f32)
  v2f a;
  a[0] = 1.0f;
  a[1] = 1.0f;
  v2f bmat;
  bmat[0] = v;
  bmat[1] = 0.0f;
  v8f c = {0.f, 0.f, 0.f, 0.f, 0.f, 0.f, 0.f, 0.f};
  c = __builtin_amdgcn_wmma_f32_16x16x4_f32(false, a, false, bmat, (short)0, c,
                                            false, false);
  float r = c[0];  // lanes 0..15: D[0][j], lanes 16..31: D[8][j]; both rows sum to total
  r += __shfl_xor(r, 1, 16);
  r += __shfl_xor(r, 2, 16);
  r += __shfl_xor(r, 4, 16);
  r += __shfl_xor(r, 8, 16);
  return r;  // every lane now holds the wave total
#else
  for (int off = 16; off >= 1; off >>= 1) v += __shfl_xor(v, off, 32);
  return v;
#endif
}

__device__ __forceinline__ void stage_prior_tile(const float* __restrict__ priors,
                                                 float* s_pri, int p, int tid) {
#if HAVE_ASYNC_LDS
  if (p < P_) {
    __builtin_amdgcn_global_load_async_to_lds_b128(
        (__attribute__((address_space(1))) v4i_ld*)(priors + (size_t)p * 4),
        (__attribute__((address_space(3))) v4i_ld*)(&s_pri[tid * 4]),
        /*offset=*/0, /*cpol=*/0);
  }
#if __has_builtin(__builtin_amdgcn_s_wait_asynccnt)
  __builtin_amdgcn_s_wait_asynccnt(0);
#else
  asm volatile("s_wait_asynccnt 0" ::: "memory");
#endif
#else
  if (p < P_) {
    const float4 t = *(const float4*)(priors + (size_t)p * 4);
    *(float4*)&s_pri[tid * 4] = t;
  }
#endif
}

__global__ __launch_bounds__(NTHREADS) void mbox_main(
    const float* __restrict__ pred_loc,  // [B,P,4]
    const float* __restrict__ pred_cls,  // [B,P,C]
    const float* __restrict__ b_boxes,   // [B,M,4] xyxy
    const int* __restrict__ b_labels,    // [B,M]
    const float* __restrict__ priors,    // [P,4] cxcywh
    float* __restrict__ part)            // [B,4]: loc_sum, conf_pos, hard_neg, n_pos
{
  const int b = blockIdx.x;
  const int tid = threadIdx.x;
  const int lane = tid & 31;
  const int wav = tid >> 5;

  __shared__ float4 s_box[M_];
  __shared__ float s_area[M_];
  __shared__ int s_label[M_];
  __shared__ int s_objidx[M_];
  __shared__ __align__(16) float s_pri[NTHREADS * 4];
  __shared__ float s_ce[P_];
  __shared__ float s_wred[NWAVES][4];
  __shared__ float s_wargv[NWAVES][M_];
  __shared__ int s_wargp[NWAVES][M_];
  __shared__ int s_bcasti;
  __shared__ int s_npos;

  if (tid < M_) {
    const float* bp = b_boxes + ((size_t)b * M_ + tid) * 4;
    float4 bb = make_float4(bp[0], bp[1], bp[2], bp[3]);
    s_box[tid] = bb;
    s_area[tid] = (bb.z - bb.x) * (bb.w - bb.y);
    s_label[tid] = b_labels[(size_t)b * M_ + tid];
  }
  __syncthreads();

  // ---------------- Pass 1: obj_idx[m] = argmax_p iou(m,p) (first-max) ------
  float bestv[M_];
  int bestp[M_];
#pragma unroll
  for (int m = 0; m < M_; ++m) {
    bestv[m] = -1.0f;
    bestp[m] = 0x7fffffff;
  }

  for (int base = 0; base < P_; base += NTHREADS) {
    __syncthreads();  // previous tile fully consumed before restaging
    const int p = base + tid;
    stage_prior_tile(priors, s_pri, p, tid);
    __syncthreads();
    if (p < P_) {
      const float cx = s_pri[tid * 4 + 0], cy = s_pri[tid * 4 + 1];
      const float pw = s_pri[tid * 4 + 2], ph = s_pri[tid * 4 + 3];
      const float px1 = cx - pw * 0.5f, py1 = cy - ph * 0.5f;
      const float px2 = cx + pw * 0.5f, py2 = cy + ph * 0.5f;
      const float pa = (px2 - px1) * (py2 - py1);
#pragma unroll
      for (int m = 0; m < M_; ++m) {
        const float4 bb = s_box[m];
        float ix = fminf(bb.z, px2) - fmaxf(bb.x, px1);
        float iy = fminf(bb.w, py2) - fmaxf(bb.y, py1);
        ix = fmaxf(ix, 0.0f);
        iy = fmaxf(iy, 0.0f);
        const float inter = ix * iy;
        const float iou = inter / (s_area[m] + pa - inter);
        if (iou > bestv[m]) {  // strict > keeps first (smallest p)
          bestv[m] = iou;
          bestp[m] = p;
        }
      }
    }
  }
  // wave-level argmax combine (tie -> smaller index), then cross-wave
#pragma unroll
  for (int m = 0; m < M_; ++m) {
    float v = bestv[m];
    int p = bestp[m];
    for (int off = 16; off >= 1; off >>= 1) {
      const float ov = __shfl_xor(v, off, 32);
      const int op = __shfl_xor(p, off, 32);
      if (ov > v || (ov == v && op < p)) {
        v = ov;
        p = op;
      }
    }
    if (lane == 0) {
      s_wargv[wav][m] = v;
      s_wargp[wav][m] = p;
    }
  }
  __syncthreads();
  if (tid < M_) {
    float v = s_wargv[0][tid];
    int p = s_wargp[0][tid];
    for (int w = 1; w < NWAVES; ++w) {
      const float ov = s_wargv[w][tid];
      const int op = s_wargp[w][tid];
      if (ov > v || (ov == v && op < p)) {
        v = ov;
        p = op;
      }
    }
    s_objidx[tid] = p;
  }
  __syncthreads();

  // ---------------- Pass 2: matching, loc L1, CE ----------------------------
  float loc_sum = 0.0f, conf_pos = 0.0f;
  int npos = 0;
  for (int base = 0; base < P_; base += NTHREADS) {
    __syncthreads();
    const int p = base + tid;
    stage_prior_tile(priors, s_pri, p, tid);
    __syncthreads();
    if (p < P_) {
      const float cx = s_pri[tid * 4 + 0], cy = s_pri[tid * 4 + 1];
      const float pw = s_pri[tid * 4 + 2], ph = s_pri[tid * 4 + 3];
      const float px1 = cx - pw * 0.5f, py1 = cy - ph * 0.5f;
      const float px2 = cx + pw * 0.5f, py2 = cy + ph * 0.5f;
      const float pa = (px2 - px1) * (py2 - py1);
      float bv = -1.0f;
      int bm = 0;
#pragma unroll
      for (int m = 0; m < M_; ++m) {
        const float4 bb = s_box[m];
        float ix = fminf(bb.z, px2) - fmaxf(bb.x, px1);
        float iy = fminf(bb.w, py2) - fmaxf(bb.y, py1);
        ix = fmaxf(ix, 0.0f);
        iy = fmaxf(iy, 0.0f);
        const float inter = ix * iy;
        float iou = inter / (s_area[m] + pa - inter);
        if (s_objidx[m] == p) iou = 1.0f;  // force-match override
        if (iou > bv) {                    // first-max semantics
          bv = iou;
          bm = m;
        }
      }
      const bool pos = (bv >= 0.5f);
      const float sgn = pos ? 1.0f : -1.0f;
      const float tcf = (float)s_label[bm] * sgn;
      const int tc = (tcf < 0.0f) ? BG_ : (int)tcf;  // label 0 stays class 0 (ref quirk)

      if (pos) {
        const float4 bb = s_box[bm];
        const float mcx = (bb.x + bb.z) * 0.5f, mcy = (bb.y + bb.w) * 0.5f;
        const float mw = bb.z - bb.x, mh = bb.w - bb.y;
        const float g0 = (mcx - cx) / (pw / 10.0f);
        const float g1 = (mcy - cy) / (ph / 10.0f);
        const float g2 = logf(mw / pw) * 5.0f;
        const float g3 = logf(mh / ph) * 5.0f;
        const float4 pl = *(const float4*)(pred_loc + ((size_t)b * P_ + p) * 4);
        loc_sum += fabsf(pl.x - g0) + fabsf(pl.y - g1) + fabsf(pl.z - g2) +
                   fabsf(pl.w - g3);
        npos++;
      }

      const float* xc = pred_cls + ((size_t)b * P_ + p) * C_;
      float mx = -INFINITY, xt = 0.0f;
#pragma unroll
      for (int i = 0; i < C_; ++i) {
        const float x = xc[i];
        mx = fmaxf(mx, x);
        if (i == tc) xt = x;
      }
      float s = 0.0f;
#pragma unroll
      for (int i = 0; i < C_; ++i) s += expf(xc[i] - mx);
      const float ce = mx + logf(s) - xt;
      s_ce[p] = pos ? 0.0f : ce;
      if (pos) conf_pos += ce;
    }
  }

  // ---------------- block reductions (WMMA wave sums) -----------------------
  const float w0 = wave_sum32(loc_sum);
  const float w1 = wave_sum32(conf_pos);
  const float w2 = wave_sum32((float)npos);
  if (lane == 0) {
    s_wred[wav][0] = w0;
    s_wred[wav][1] = w1;
    s_wred[wav][2] = w2;
  }
  __syncthreads();
  float blk_loc = 0.0f, blk_conf = 0.0f;
  if (tid == 0) {
    float a = 0.0f, c2 = 0.0f, n = 0.0f;
    for (int w = 0; w < NWAVES; ++w) {
      a += s_wred[w][0];
      c2 += s_wred[w][1];
      n += s_wred[w][2];
    }
    blk_loc = a;
    blk_conf = c2;
    s_npos = (int)(n + 0.5f);
  }
  __syncthreads();
  const int nposb = s_npos;

  // ---------------- hard-negative top-K sum via bit-pattern select ----------
  float hard = 0.0f;
  long long Kl = 3LL * (long long)nposb;
  const int K = (Kl > (long long)P_) ? P_ : (int)Kl;
  if (K > 0) {
    // v_K = K-th largest of s_ce (all values >= 0); binary search max T with
    // count(x >= T) >= K over non-negative float bit patterns.
    unsigned lo = 0u, hi = 0x7F800000u;
    while (lo < hi) {
      const unsigned mid = lo + ((hi - lo + 1u) >> 1);
      const float T = __uint_as_float(mid);
      int cnt = 0;
      for (int p = tid; p < P_; p += NTHREADS) cnt += (s_ce[p] >= T) ? 1 : 0;
      const float cf = wave_sum32((float)cnt);
      if (lane == 0) s_wred[wav][3] = cf;
      __syncthreads();
      if (tid == 0) {
        float t = 0.0f;
        for (int w = 0; w < NWAVES; ++w) t += s_wred[w][3];
        s_bcasti = (int)(t + 0.5f);
      }
      __syncthreads();
      const int total = s_bcasti;
      if (total >= K)
        lo = mid;
      else
        hi = mid - 1u;
    }
    const float vK = __uint_as_float(lo);
    float sgt = 0.0f;
    int cgt = 0;
    for (int p = tid; p < P_; p += NTHREADS) {
      const float v = s_ce[p];
      if (v > vK) {
        sgt += v;
        cgt++;
      }
    }
    const float sg = wave_sum32(sgt);
    const float cg = wave_sum32((float)cgt);
    if (lane == 0) {
      s_wred[wav][0] = sg;
      s_wred[wav][1] = cg;
    }
    __syncthreads();
    if (tid == 0) {
      float S = 0.0f, Cg = 0.0f;
      for (int w = 0; w < NWAVES; ++w) {
        S += s_wred[w][0];
        Cg += s_wred[w][1];
      }
      hard = S + ((float)K - Cg) * vK;  // == sum of top-K (ties included)
    }
  }

  if (tid == 0) {
    part[b * 4 + 0] = blk_loc;
    part[b * 4 + 1] = blk_conf;
    part[b * 4 + 2] = hard;
    part[b * 4 + 3] = (float)nposb;
  }
}

__global__ void mbox_final(const float* __restrict__ part, float* __restrict__ out) {
  if (blockIdx.x == 0 && threadIdx.x == 0) {
    float loc = 0.0f, cp = 0.0f, hn = 0.0f, np = 0.0f;
    for (int b = 0; b < B_; ++b) {  // fixed order: deterministic
      loc += part[b * 4 + 0];
      cp += part[b * 4 + 1];
      hn += part[b * 4 + 2];
      np += part[b * 4 + 3];
    }
    float loc_loss = loc / (np * 4.0f);
    const float conf_loss = (hn + cp) / np;
    loc_loss = 1.0f * loc_loss;  // ALPHA
    out[0] = conf_loss + loc_loss;
    out[1] = loc_loss;
    out[2] = conf_loss;
  }
}

extern "C" void kernel_launch(void* const* d_in, const int* in_sizes, int n_in,
                              void* d_out, int out_size, void* d_ws, size_t ws_size,
                              hipStream_t stream) {
  (void)in_sizes;
  (void)n_in;
  (void)out_size;
  (void)ws_size;
  const float* pred_loc = (const float*)d_in[0];
  const float* pred_cls = (const float*)d_in[1];
  const float* b_boxes = (const float*)d_in[2];
  const int* b_labels = (const int*)d_in[3];
  const float* priors = (const float*)d_in[4];
  float* part = (float*)d_ws;  // B*4 floats

  mbox_main<<<B_, NTHREADS, 0, stream>>>(pred_loc, pred_cls, b_boxes, b_labels,
                                         priors, part);
  mbox_final<<<1, 64, 0, stream>>>(part, (float*)d_out);
}